// FakeProfile_16183436772069
// MI455X (gfx1250) — compile-verified
//
#include <hip/hip_runtime.h>
#include <stdint.h>

#define ROWS      2048
#define COLS      20000
#define KSEL      64
#define THRESH    0.5f
#define CHUNK     128                 // elements per staged chunk (32 lanes x 4)
#define NFULL     (COLS / CHUNK)      // 156 full chunks
#define TAIL_BASE (NFULL * CHUNK)     // 19968; tail = 32 elements

typedef float v4f __attribute__((ext_vector_type(4)));
typedef int   v4i __attribute__((vector_size(16)));   // builtin's pointee type

#define AS1 __attribute__((address_space(1)))
#define AS3 __attribute__((address_space(3)))

#if __has_builtin(__builtin_amdgcn_global_load_async_to_lds_b128)
#define USE_ASYNC 1
#else
#define USE_ASYNC 0
#endif

#if USE_ASYNC
#if __has_builtin(__builtin_amdgcn_s_wait_asynccnt)
#define WAIT_ASYNC(n) __builtin_amdgcn_s_wait_asynccnt(n)
#else
#define WAIT_ASYNC(n) asm volatile("s_wait_asynccnt %0" ::"i"(n) : "memory")
#endif

// One b128 async copy per lane: 32 lanes x 16B = 512B = 128 floats per call.
__device__ __forceinline__ void stage_chunk(const float* g_in, const float* g_pr,
                                            float* l_in, float* l_pr, int lane) {
  const float* gi = g_in + lane * 4;
  const float* gp = g_pr + lane * 4;
  float*       li = l_in + lane * 4;
  float*       lp = l_pr + lane * 4;
  __builtin_amdgcn_global_load_async_to_lds_b128(
      (AS1 v4i*)(uintptr_t)gi, (AS3 v4i*)(uintptr_t)li, 0, 0);
  __builtin_amdgcn_global_load_async_to_lds_b128(
      (AS1 v4i*)(uintptr_t)gp, (AS3 v4i*)(uintptr_t)lp, 0, 0);
}
#endif

// Rank the ones of a 128-element chunk (element order: j = base + k*32 + lane).
// If the globally-64th one is in this chunk, its lane records J in *sJ.
__device__ __forceinline__ int rank_scan(bool o0, bool o1, bool o2, bool o3,
                                         int lane, int cnt, int base, int* sJ) {
  unsigned m0 = __builtin_amdgcn_ballot_w32(o0);
  unsigned m1 = __builtin_amdgcn_ballot_w32(o1);
  unsigned m2 = __builtin_amdgcn_ballot_w32(o2);
  unsigned m3 = __builtin_amdgcn_ballot_w32(o3);
  int t0 = __popc(m0), t1 = __popc(m1), t2 = __popc(m2), t3 = __popc(m3);
  int total = t0 + t1 + t2 + t3;
  if (cnt + total >= KSEL) {                 // the 64th one crosses in this chunk
    unsigned lt = (1u << lane) - 1u;
    int r0 = cnt + __popc(m0 & lt);
    int r1 = cnt + t0 + __popc(m1 & lt);
    int r2 = cnt + t0 + t1 + __popc(m2 & lt);
    int r3 = cnt + t0 + t1 + t2 + __popc(m3 & lt);
    if (o0 && r0 == KSEL - 1) *sJ = base + 0 * 32 + lane;
    if (o1 && r1 == KSEL - 1) *sJ = base + 1 * 32 + lane;
    if (o2 && r2 == KSEL - 1) *sJ = base + 2 * 32 + lane;
    if (o3 && r3 == KSEL - 1) *sJ = base + 3 * 32 + lane;
  }
  return cnt + total;
}

__global__ void __launch_bounds__(256)
first64_ones_kernel(const float* __restrict__ param,
                    const float* __restrict__ inp,
                    float* __restrict__ out) {
  __shared__ int   sJ;
  __shared__ float sIn[2][CHUNK];
  __shared__ float sPr[2][CHUNK];

  const int row = blockIdx.x;
  const int tid = threadIdx.x;
  const float* prow = param + (size_t)row * COLS;
  const float* irow = inp   + (size_t)row * COLS;
  float*       orow = out   + (size_t)row * COLS;

  // ---------- Phase A: wave 0 finds J = column of the 64th one ----------
  if (tid < 32) {
    const int lane = tid;
    if (lane == 0) sJ = COLS;   // sentinel: fewer than 64 ones -> keep all
    int cnt = 0;

#if USE_ASYNC
    stage_chunk(irow, prow, sIn[0], sPr[0], lane);
    for (int c = 0; c < NFULL; ++c) {
      const int buf = c & 1;
      if (c + 1 < NFULL) {
        stage_chunk(irow + (c + 1) * CHUNK, prow + (c + 1) * CHUNK,
                    sIn[buf ^ 1], sPr[buf ^ 1], lane);
        WAIT_ASYNC(2);          // chunk c's 2 async ops complete (in-order)
      } else {
        WAIT_ASYNC(0);
      }
      asm volatile("" ::: "memory");   // keep LDS reads below the wait
      bool o0 = sIn[buf][lane +  0] > 0.f && sPr[buf][lane +  0] > THRESH;
      bool o1 = sIn[buf][lane + 32] > 0.f && sPr[buf][lane + 32] > THRESH;
      bool o2 = sIn[buf][lane + 64] > 0.f && sPr[buf][lane + 64] > THRESH;
      bool o3 = sIn[buf][lane + 96] > 0.f && sPr[buf][lane + 96] > THRESH;
      cnt = rank_scan(o0, o1, o2, o3, lane, cnt, c * CHUNK, &sJ);
      if (cnt >= KSEL) break;   // leftover asyncs drain at S_ENDPGM's wait-idle
    }
#else
    for (int c = 0; c < NFULL; ++c) {
      const int base = c * CHUNK;
      if (c + 2 < NFULL) {      // gfx1250 global_prefetch_b8 two chunks ahead
        __builtin_prefetch(irow + base + 2 * CHUNK + lane * 4, 0, 0);
        __builtin_prefetch(prow + base + 2 * CHUNK + lane * 4, 0, 0);
      }
      bool o0 = irow[base + lane +  0] > 0.f && prow[base + lane +  0] > THRESH;
      bool o1 = irow[base + lane + 32] > 0.f && prow[base + lane + 32] > THRESH;
      bool o2 = irow[base + lane + 64] > 0.f && prow[base + lane + 64] > THRESH;
      bool o3 = irow[base + lane + 96] > 0.f && prow[base + lane + 96] > THRESH;
      cnt = rank_scan(o0, o1, o2, o3, lane, cnt, base, &sJ);
      if (cnt >= KSEL) break;
    }
#endif

    if (cnt < KSEL) {           // tail: last 32 elements (COLS % 128 == 32)
      bool o = irow[TAIL_BASE + lane] > 0.f && prow[TAIL_BASE + lane] > THRESH;
      unsigned m  = __builtin_amdgcn_ballot_w32(o);
      unsigned lt = (1u << lane) - 1u;
      int r = cnt + __popc(m & lt);
      if (o && r == KSEL - 1) sJ = TAIL_BASE + lane;
    }
  }
  __syncthreads();
  const int J = sJ;

  // ---------- Phase B: write whole row once, 128-bit NT stores ----------
  const v4f* __restrict__ i4 = (const v4f*)irow;
  const v4f* __restrict__ p4 = (const v4f*)prow;
  v4f* o4 = (v4f*)orow;
  for (int g = tid; g < COLS / 4; g += 256) {
    const int j0 = g * 4;
    v4f r;
    if (j0 > J) {               // past cutoff: pure store, no loads (common case)
      r = (v4f){0.f, 0.f, 0.f, 0.f};
    } else {
      v4f iv = i4[g];
      v4f pv = p4[g];
      r.x = (iv.x > 0.f && pv.x > THRESH && (j0 + 0) <= J) ? 1.f : 0.f;
      r.y = (iv.y > 0.f && pv.y > THRESH && (j0 + 1) <= J) ? 1.f : 0.f;
      r.z = (iv.z > 0.f && pv.z > THRESH && (j0 + 2) <= J) ? 1.f : 0.f;
      r.w = (iv.w > 0.f && pv.w > THRESH && (j0 + 3) <= J) ? 1.f : 0.f;
    }
    __builtin_nontemporal_store(r, &o4[g]);   // output is write-once: bypass cache
  }
}

extern "C" void kernel_launch(void* const* d_in, const int* in_sizes, int n_in,
                              void* d_out, int out_size, void* d_ws, size_t ws_size,
                              hipStream_t stream) {
  const float* param = (const float*)d_in[0];   // fake_parameter [2048,20000]
  const float* inp   = (const float*)d_in[1];   // input          [2048,20000]
  float* out = (float*)d_out;
  (void)in_sizes; (void)n_in; (void)out_size; (void)d_ws; (void)ws_size;
  hipLaunchKernelGGL(first64_ones_kernel, dim3(ROWS), dim3(256), 0, stream,
                     param, inp, out);
}